// HeadVer1_14035953123646
// MI455X (gfx1250) — compile-verified
//
#include <hip/hip_runtime.h>

// Problem shape (fixed by the reference): x is (B, T, C) float32.
constexpr int B = 8;
constexpr int T = 4096;
constexpr int C = 1024;

constexpr int NCHUNK = 64;          // chunks along T
constexpr int CHUNK  = T / NCHUNK;  // 64 timesteps per chunk
constexpr int C4     = C / 4;       // 256 float4 columns -> one block covers all of C

constexpr int TS = 4;               // timesteps per LDS tile (phase 3)
constexpr int NT = CHUNK / TS;      // 16 tiles per chunk

// ---------------------------------------------------------------------------
// Phase 1: per-(b, chunk, c4) partial sums of x over the chunk's timesteps.
// ws layout: float4 ws4[B][NCHUNK][C4]  (2 MiB total, lives in L2)
// Direct register streaming: compiler emits s_clause'd global_load_b128 +
// v_pk_add_f32, which saturates HBM.
// ---------------------------------------------------------------------------
__global__ __launch_bounds__(C4) void chunk_sums(const float* __restrict__ x,
                                                 float* __restrict__ ws) {
    const int blk   = blockIdx.x;        // b * NCHUNK + chunk
    const int b     = blk / NCHUNK;
    const int chunk = blk % NCHUNK;
    const int c4    = threadIdx.x;       // 0..C4-1

    const float4* __restrict__ xp =
        reinterpret_cast<const float4*>(x) +
        ((size_t)(b * T + chunk * CHUNK) * C4 + c4);

    float4 acc = make_float4(0.f, 0.f, 0.f, 0.f);
#pragma unroll 8
    for (int i = 0; i < CHUNK; ++i) {
        if (i + 16 < CHUNK) {
            // gfx1250: global_prefetch_b8 — pull the stream 16 steps ahead
            __builtin_prefetch((const void*)(xp + (size_t)(i + 16) * C4), 0, 3);
        }
        float4 v = xp[(size_t)i * C4];
        acc.x += v.x; acc.y += v.y; acc.z += v.z; acc.w += v.w;
    }

    reinterpret_cast<float4*>(ws)[(size_t)(b * NCHUNK + chunk) * C4 + c4] = acc;
}

// ---------------------------------------------------------------------------
// Phase 2: in-place exclusive prefix over the NCHUNK partial sums, per (b,c4).
// Only 2 MiB touched; pure L2 traffic.
// ---------------------------------------------------------------------------
__global__ __launch_bounds__(256) void exclusive_chunk_prefix(float* __restrict__ ws) {
    const int tid = blockIdx.x * blockDim.x + threadIdx.x;  // 0 .. B*C4-1
    if (tid >= B * C4) return;
    const int b  = tid / C4;
    const int c4 = tid % C4;

    float4* __restrict__ p =
        reinterpret_cast<float4*>(ws) + ((size_t)b * NCHUNK) * C4 + c4;

    float4 run = make_float4(0.f, 0.f, 0.f, 0.f);
#pragma unroll 4
    for (int k = 0; k < NCHUNK; ++k) {
        float4 v = p[(size_t)k * C4];
        p[(size_t)k * C4] = run;
        run.x += v.x; run.y += v.y; run.z += v.z; run.w += v.w;
    }
}

// ---------------------------------------------------------------------------
// Phase 3: re-scan each chunk with its exclusive base, divide by (t+1), store.
// gfx1250 async path: tiles of x are staged global->LDS with
// GLOBAL_LOAD_ASYNC_TO_LDS_B128 (ASYNCcnt), double-buffered so tile k+1 is in
// flight while tile k is consumed from LDS. Consumption is lane-local (each
// thread reads back exactly the column it loaded), so no barriers are needed;
// async loads complete in order, so count-based s_wait_asynccnt is tile-exact.
// ---------------------------------------------------------------------------
__global__ __launch_bounds__(C4) void scan_out(const float* __restrict__ x,
                                               const float* __restrict__ ws,
                                               float* __restrict__ out) {
    __shared__ float4 smem[2][TS][C4];   // 2 x 16 KiB double buffer

    const int blk   = blockIdx.x;        // b * NCHUNK + chunk
    const int b     = blk / NCHUNK;
    const int chunk = blk % NCHUNK;
    const int c4    = threadIdx.x;

    const size_t col_base = (size_t)(b * T + chunk * CHUNK) * C4 + c4;
    const float4* __restrict__ xp = reinterpret_cast<const float4*>(x) + col_base;
    float4*       __restrict__ op = reinterpret_cast<float4*>(out)     + col_base;

    // Issue one tile's worth of async global->LDS loads (TS per thread).
    auto issue_tile = [&](int buf, int tile) {
#pragma unroll
        for (int r = 0; r < TS; ++r) {
            // ISA §10.2: LDS aperture uses addr[31:0] as the wave-relative LDS
            // byte address, matching the compiler's ds offsets for smem.
            unsigned lds_addr = (unsigned)(size_t)&smem[buf][r][c4];
            unsigned long long gaddr =
                (unsigned long long)(size_t)(xp + (size_t)(tile * TS + r) * C4);
            asm volatile("global_load_async_to_lds_b128 %0, %1, off"
                         :: "v"(lds_addr), "v"(gaddr)
                         : "memory");
        }
    };

    issue_tile(0, 0);

    float4 acc =
        reinterpret_cast<const float4*>(ws)[(size_t)(b * NCHUNK + chunk) * C4 + c4];
    const int t0 = chunk * CHUNK;

    for (int tile = 0; tile < NT; ++tile) {
        const int buf = tile & 1;
        if (tile + 1 < NT) {
            // Close the LDS WAR window on the buffer we are about to overwrite.
            asm volatile("s_wait_dscnt 0" ::: "memory");
            issue_tile(buf ^ 1, tile + 1);
            // Allow only the next tile's TS loads to remain outstanding:
            // current tile's data is then guaranteed resident in LDS.
            asm volatile("s_wait_asynccnt %0" :: "i"(TS) : "memory");
        } else {
            asm volatile("s_wait_asynccnt 0" ::: "memory");
        }

#pragma unroll
        for (int r = 0; r < TS; ++r) {
            float4 v = smem[buf][r][c4];     // ds_load_b128, lane-local column
            acc.x += v.x; acc.y += v.y; acc.z += v.z; acc.w += v.w;

            const float denom = (float)(t0 + tile * TS + r + 1);
            float4 o;
            o.x = acc.x / denom;
            o.y = acc.y / denom;
            o.z = acc.z / denom;
            o.w = acc.w / denom;
            op[(size_t)(tile * TS + r) * C4] = o;
        }
    }
}

// ---------------------------------------------------------------------------
extern "C" void kernel_launch(void* const* d_in, const int* in_sizes, int n_in,
                              void* d_out, int out_size, void* d_ws, size_t ws_size,
                              hipStream_t stream) {
    (void)in_sizes; (void)n_in; (void)out_size; (void)ws_size;

    const float* x   = (const float*)d_in[0];
    float*       out = (float*)d_out;
    float*       ws  = (float*)d_ws;   // needs B*NCHUNK*C floats = 2 MiB

    // Phase 1: partial sums per chunk (streams x from HBM, warms L2)
    chunk_sums<<<B * NCHUNK, C4, 0, stream>>>(x, ws);

    // Phase 2: exclusive prefix over chunk sums (tiny, L2-resident)
    exclusive_chunk_prefix<<<(B * C4 + 255) / 256, 256, 0, stream>>>(ws);

    // Phase 3: async-LDS double-buffered scan + divide (x re-read from L2)
    scan_out<<<B * NCHUNK, C4, 0, stream>>>(x, ws, out);
}